// myModel_39402029973982
// MI455X (gfx1250) — compile-verified
//
#include <hip/hip_runtime.h>
#include <hip/hip_bf16.h>
#include <math.h>

#define DIM 64

typedef __attribute__((ext_vector_type(2))) float v2f;
typedef __attribute__((ext_vector_type(8))) float v8f;

// ---------------------------------------------------------------- zero fill
__global__ void k_zero(float* __restrict__ p, long long n) {
  long long i = (long long)blockIdx.x * blockDim.x + threadIdx.x;
  long long s = (long long)gridDim.x * blockDim.x;
  for (; i < n; i += s) p[i] = 0.0f;
}

// ---------------------------------------------------- fused SpMM + SpMM^T
// One edge per 16-lane group; each lane owns 4 consecutive d's (float4).
//   ue[row] += val * xi[col]   (forward spmm,  out [U,64])
//   ie[col] += val * xu[row]   (transpose spmm, out [I,64])
// Indices read once per edge set; scatters are L2-resident f32 atomics.
__global__ void k_spmm_fused(const int* __restrict__ rows,
                             const int* __restrict__ cols,
                             const float* __restrict__ vals,
                             const float* __restrict__ xi,   // [I,64]
                             const float* __restrict__ xu,   // [U,64]
                             float* __restrict__ ue,         // [U,64]
                             float* __restrict__ ie,         // [I,64]
                             int E) {
  long long t = (long long)blockIdx.x * blockDim.x + threadIdx.x;
  long long estride = ((long long)gridDim.x * blockDim.x) >> 4;
  int dbase = (int)(t & 15) * 4;
  for (long long e = (t >> 4); e < E; e += estride) {
    if (e + estride < E) {  // stream-ahead on the sequential index arrays
      __builtin_prefetch(rows + e + estride, 0, 1);
      __builtin_prefetch(cols + e + estride, 0, 1);
      __builtin_prefetch(vals + e + estride, 0, 1);
    }
    int r = rows[e];
    int c = cols[e];
    float v = vals[e];
    float4 a = *(const float4*)(xi + (long long)c * DIM + dbase);
    float4 b = *(const float4*)(xu + (long long)r * DIM + dbase);
    float* up = ue + (long long)r * DIM + dbase;
    float* ip = ie + (long long)c * DIM + dbase;
    atomicAdd(up + 0, v * a.x);
    atomicAdd(up + 1, v * a.y);
    atomicAdd(up + 2, v * a.z);
    atomicAdd(up + 3, v * a.w);
    atomicAdd(ip + 0, v * b.x);
    atomicAdd(ip + 1, v * b.y);
    atomicAdd(ip + 2, v * b.z);
    atomicAdd(ip + 3, v * b.w);
  }
}

// --------------------------------------- sigmoid(mean(e0..e3) @ W), WMMA f32
// One wave -> one 16x16 output tile. 4 waves/block cover 64 columns.
// K = 64 -> 16 x V_WMMA_F32_16X16X4_F32 accumulation steps.
__global__ void k_mean_gemm_sigmoid(const float* __restrict__ e0,
                                    const float* __restrict__ e1,
                                    const float* __restrict__ e2,
                                    const float* __restrict__ e3,
                                    const float* __restrict__ w,  // [64,64] (K,N) row-major
                                    float* __restrict__ out,      // [N,64]
                                    int N) {
  int wave = threadIdx.x >> 5;        // 0..3 -> column tile
  int lane = threadIdx.x & 31;
  int half = lane & 15;
  int hi   = lane >> 4;               // 0: K{0,1}/M rows 0-7 of D; 1: K{2,3}/M rows 8-15
  int rowTile = blockIdx.x;

  int arow = rowTile * 16 + half;     // A-matrix row for this lane (both halves)
  if (arow >= N) return;
  long long abase = (long long)arow * DIM;
  int bn = wave * 16 + half;          // B/N column for this lane

  v8f c = {};
  #pragma unroll
  for (int kk = 0; kk < 16; ++kk) {
    int k0 = kk * 4 + hi * 2;         // this lane's K pair per ISA 16x4 f32 layout
    v2f a, b;
    a.x = 0.25f * (e0[abase + k0]     + e1[abase + k0]     + e2[abase + k0]     + e3[abase + k0]);
    a.y = 0.25f * (e0[abase + k0 + 1] + e1[abase + k0 + 1] + e2[abase + k0 + 1] + e3[abase + k0 + 1]);
    b.x = w[(long long)k0 * DIM + bn];
    b.y = w[(long long)(k0 + 1) * DIM + bn];
    c = __builtin_amdgcn_wmma_f32_16x16x4_f32(false, a, false, b, (short)0, c,
                                              false, false);
  }
  // D layout: lanes 0-15 -> M = r, lanes 16-31 -> M = 8 + r; N = half (+tile)
  int orow0 = rowTile * 16 + hi * 8;
  int ocol  = wave * 16 + half;
  #pragma unroll
  for (int r = 0; r < 8; ++r) {
    float x = c[r];
    out[(long long)(orow0 + r) * DIM + ocol] = 1.0f / (1.0f + __expf(-x));
  }
}

// -------------------------------- column sum-of-squares over dim=1 (rows)
// norm2[k*64+d] += sum_u e_k[u][d]^2   (LDS partial, 4 atomics/(d) per block)
__global__ void k_colnorm2(const float* __restrict__ e0,
                           const float* __restrict__ e1,
                           const float* __restrict__ e2,
                           const float* __restrict__ e3,
                           float* __restrict__ norm2,  // [4*64]
                           int N) {
  __shared__ float lds[8][4][DIM];
  int d = threadIdx.x;   // 0..63
  int ty = threadIdx.y;  // 0..7
  float s0 = 0.f, s1 = 0.f, s2 = 0.f, s3 = 0.f;
  for (long long u = (long long)blockIdx.x * 8 + ty; u < N;
       u += (long long)gridDim.x * 8) {
    long long idx = u * DIM + d;
    float a = e0[idx]; s0 += a * a;
    float b = e1[idx]; s1 += b * b;
    float c = e2[idx]; s2 += c * c;
    float f = e3[idx]; s3 += f * f;
  }
  lds[ty][0][d] = s0;
  lds[ty][1][d] = s1;
  lds[ty][2][d] = s2;
  lds[ty][3][d] = s3;
  __syncthreads();
  if (ty < 4) {
    float s = 0.f;
    #pragma unroll
    for (int j = 0; j < 8; ++j) s += lds[j][ty][d];
    atomicAdd(&norm2[ty * DIM + d], s);
  }
}

// ------------------------- write normalized stack: outn[k][u][d] = e/max(||col||,eps)
__global__ void k_normalize(const float* __restrict__ e0,
                            const float* __restrict__ e1,
                            const float* __restrict__ e2,
                            const float* __restrict__ e3,
                            const float* __restrict__ norm2,  // [4*64]
                            float* __restrict__ outn,         // [4,N,64]
                            int N) {
  const float* es[4] = {e0, e1, e2, e3};
  long long nq = (long long)4 * N * (DIM / 4);  // float4 work items
  long long t = (long long)blockIdx.x * blockDim.x + threadIdx.x;
  long long stride = (long long)gridDim.x * blockDim.x;
  long long plane = (long long)N * DIM;
  for (; t < nq; t += stride) {
    long long idx = t * 4;
    int k = (int)(idx / plane);
    long long rem = idx - (long long)k * plane;
    int d = (int)(rem & (DIM - 1));
    float4 v = *(const float4*)(es[k] + rem);
    float4 r;
    r.x = v.x / fmaxf(sqrtf(norm2[k * DIM + d + 0]), 1e-12f);
    r.y = v.y / fmaxf(sqrtf(norm2[k * DIM + d + 1]), 1e-12f);
    r.z = v.z / fmaxf(sqrtf(norm2[k * DIM + d + 2]), 1e-12f);
    r.w = v.w / fmaxf(sqrtf(norm2[k * DIM + d + 3]), 1e-12f);
    *(float4*)(outn + idx) = r;
  }
}

extern "C" void kernel_launch(void* const* d_in, const int* in_sizes, int n_in,
                              void* d_out, int out_size, void* d_ws, size_t ws_size,
                              hipStream_t stream) {
  (void)n_in; (void)out_size; (void)ws_size;
  const float* user_emb = (const float*)d_in[0];
  const float* item_emb = (const float*)d_in[1];
  const float* uu0 = (const float*)d_in[2];
  const float* ii0 = (const float*)d_in[3];
  const float* uu1 = (const float*)d_in[4];
  const float* ii1 = (const float*)d_in[5];
  const float* uu2 = (const float*)d_in[6];
  const float* ii2 = (const float*)d_in[7];
  const float* u_w = (const float*)d_in[8];
  const float* i_w = (const float*)d_in[9];

  const int U = in_sizes[0] / DIM;
  const int I = in_sizes[1] / DIM;
  const int E = in_sizes[10];
  const long long UD = (long long)U * DIM;
  const long long ID = (long long)I * DIM;

  // d_out layout (return order, flat):
  float* out      = (float*)d_out;
  float* user_out = out;
  float* item_out = user_out + UD;
  float* uen      = item_out + ID;        // [4,U,64]
  float* ien      = uen + 4 * UD;         // [4,I,64]
  float* ue0      = ien + 4 * ID;
  float* ie0      = ue0 + UD;
  float* ue1      = ie0 + ID;
  float* ie1      = ue1 + UD;
  float* ue2      = ie1 + ID;
  float* ie2      = ue2 + UD;

  float* ws  = (float*)d_ws;
  float* ue3 = ws;                        // [U,64]
  float* ie3 = ue3 + UD;                  // [I,64]
  float* n2u = ie3 + ID;                  // [4*64]
  float* n2i = n2u + 4 * DIM;             // [4*64]

  // 1) zero accumulators + norm buffers (outputs are poisoned each run)
  k_zero<<<2048, 256, 0, stream>>>(ue0, UD);
  k_zero<<<2048, 256, 0, stream>>>(ue1, UD);
  k_zero<<<2048, 256, 0, stream>>>(ue2, UD);
  k_zero<<<2048, 256, 0, stream>>>(ie0, ID);
  k_zero<<<2048, 256, 0, stream>>>(ie1, ID);
  k_zero<<<2048, 256, 0, stream>>>(ie2, ID);
  k_zero<<<2048, 256, 0, stream>>>(ue3, UD);
  k_zero<<<2048, 256, 0, stream>>>(ie3, ID);
  k_zero<<<1, 256, 0, stream>>>(n2u, 4 * DIM);
  k_zero<<<1, 256, 0, stream>>>(n2i, 4 * DIM);

  // 2) fused forward + transpose SpMM per edge set
  long long spmm_threads = (long long)E * 16;
  int spmm_blocks = (int)((spmm_threads + 255) / 256);
  const int* rows[4] = {(const int*)d_in[10], (const int*)d_in[13],
                        (const int*)d_in[16], (const int*)d_in[19]};
  const int* cols[4] = {(const int*)d_in[11], (const int*)d_in[14],
                        (const int*)d_in[17], (const int*)d_in[20]};
  const float* vals[4] = {(const float*)d_in[12], (const float*)d_in[15],
                          (const float*)d_in[18], (const float*)d_in[21]};
  const float* xi[4] = {ii0, ii1, ii2, item_emb};
  const float* xu[4] = {uu0, uu1, uu2, user_emb};
  float* ues[4] = {ue0, ue1, ue2, ue3};
  float* ies[4] = {ie0, ie1, ie2, ie3};
  for (int k = 0; k < 4; ++k) {
    k_spmm_fused<<<spmm_blocks, 256, 0, stream>>>(
        rows[k], cols[k], vals[k], xi[k], xu[k], ues[k], ies[k], E);
  }

  // 3) sigmoid(mean @ W) via fp32 WMMA (16x16 tile per wave, 4 waves/block)
  k_mean_gemm_sigmoid<<<(U + 15) / 16, 128, 0, stream>>>(
      ue0, ue1, ue2, ue3, u_w, user_out, U);
  k_mean_gemm_sigmoid<<<(I + 15) / 16, 128, 0, stream>>>(
      ie0, ie1, ie2, ie3, i_w, item_out, I);

  // 4) column L2 norms over dim=1, then normalized stacks
  k_colnorm2<<<512, dim3(DIM, 8), 0, stream>>>(ue0, ue1, ue2, ue3, n2u, U);
  k_colnorm2<<<512, dim3(DIM, 8), 0, stream>>>(ie0, ie1, ie2, ie3, n2i, I);

  long long nqU = (long long)4 * U * (DIM / 4);
  long long nqI = (long long)4 * I * (DIM / 4);
  k_normalize<<<(int)((nqU + 255) / 256), 256, 0, stream>>>(
      ue0, ue1, ue2, ue3, n2u, uen, U);
  k_normalize<<<(int)((nqI + 255) / 256), 256, 0, stream>>>(
      ie0, ie1, ie2, ie3, n2i, ien, I);
}